// Dense3DSpatialTransformer_22600117912000
// MI455X (gfx1250) — compile-verified
//
#include <hip/hip_runtime.h>

// Fixed problem shape from the reference.
#define DD 160
#define HH 192
#define WW 224
constexpr unsigned HWC  = HH * WW;        // 43008
constexpr unsigned NVOX = DD * HH * WW;   // 6,881,280

typedef __attribute__((ext_vector_type(4))) float f32x4;

// Per-axis trilinear helper against a zero-padded volume (pad = 1 each side),
// matching the reference: i = floor(pos); floor corner = padded coord i,
// ceil corner = padded coord i+1; weights wf = (i+1) - pos, wc = 1 - wf.
// The reference's clip of indices to [0, dim+1] only alters the weight when
// BOTH corners fall in the zero pad (contribution is 0 regardless), so the
// weights can be computed unclipped. Pad-validity is folded into the weights
// (cmp+cndmask); offsets are clamped in-range (v_med3) so all gathers are
// unconditional, and carried as UNSIGNED so the backend can use the saddr
// global-load form (SGPR base + zext 32-bit VGPR offset) with no 64-bit
// per-gather address arithmetic.
__device__ __forceinline__ void axis(float pos, int dim, unsigned stride,
                                     unsigned& of, unsigned& oc,
                                     float& wf, float& wc)
{
    const float f   = floorf(pos);
    const int   i   = (int)f;
    const float wfu = (f + 1.0f) - pos;   // floor-corner weight (reference formula)
    const float wcu = 1.0f - wfu;         // ceil-corner weight
    wf = ((unsigned)(i - 1) < (unsigned)dim) ? wfu : 0.0f;  // valid iff 1<=i<=dim
    wc = ((unsigned)(i)     < (unsigned)dim) ? wcu : 0.0f;  // valid iff 1<=i+1<=dim
    of = (unsigned)min(max(i - 1, 0), dim - 1) * stride;    // 0-based image coord
    oc = (unsigned)min(max(i,     0), dim - 1) * stride;
}

__global__ __launch_bounds__(224) void dense3d_warp_kernel(
    const float* __restrict__ img,   // [DD, HH, WW]
    const float* __restrict__ dDp,   // displacement plane 0
    const float* __restrict__ dHp,   // displacement plane 1
    const float* __restrict__ dWp,   // displacement plane 2
    float* __restrict__ out)         // [DD, HH, WW]
{
    // block = (56, 4): 56 vec4 columns cover W=224 exactly; 4 rows per block.
    // 224 threads = exactly 7 full wave32s; no div/mod needed anywhere.
    const unsigned w0   = threadIdx.x * 4u;
    const unsigned h    = blockIdx.y * 4u + threadIdx.y;
    const unsigned d    = blockIdx.z;
    const unsigned base = (d * HH + h) * WW + w0;

    // Read-once displacement streams: non-temporal so they don't evict the
    // reused image from WGP$/L2.
    const f32x4 dDv = __builtin_nontemporal_load((const f32x4*)(dDp + base));
    const f32x4 dHv = __builtin_nontemporal_load((const f32x4*)(dHp + base));
    const f32x4 dWv = __builtin_nontemporal_load((const f32x4*)(dWp + base));

    // Pull the streams ~32KB ahead (global_prefetch_b8; speculative, OOB dropped).
    constexpr unsigned PF = 8192;
    __builtin_prefetch(dDp + base + PF, 0, 0);
    __builtin_prefetch(dHp + base + PF, 0, 0);
    __builtin_prefetch(dWp + base + PF, 0, 0);

    const float dpos = (float)(d + 1);   // padded-coordinate positions
    const float hpos = (float)(h + 1);

    f32x4 res;
#pragma unroll
    for (int j = 0; j < 4; ++j) {
        unsigned odf, odc, ohf, ohc, owf, owc;
        float    wdf, wdc, whf, whc, wwf, wwc;
        axis(dDv[j] + dpos,                 DD, HWC, odf, odc, wdf, wdc);
        axis(dHv[j] + hpos,                 HH, WW,  ohf, ohc, whf, whc);
        axis(dWv[j] + (float)(w0 + j + 1),  WW, 1u,  owf, owc, wwf, wwc);

        const unsigned r00 = odf + ohf, r01 = odf + ohc;
        const unsigned r10 = odc + ohf, r11 = odc + ohc;

        // 8 unconditional regular-temporal gathers (image reused ~8x, fits L2).
        const float v000 = img[r00 + owf], v001 = img[r00 + owc];
        const float v010 = img[r01 + owf], v011 = img[r01 + owc];
        const float v100 = img[r10 + owf], v101 = img[r10 + owc];
        const float v110 = img[r11 + owf], v111 = img[r11 + owc];

        res[j] = wdf * (whf * (wwf * v000 + wwc * v001) +
                        whc * (wwf * v010 + wwc * v011)) +
                 wdc * (whf * (wwf * v100 + wwc * v101) +
                        whc * (wwf * v110 + wwc * v111));
    }

    // Write-once output: non-temporal store.
    __builtin_nontemporal_store(res, (f32x4*)(out + base));
}

extern "C" void kernel_launch(void* const* d_in, const int* in_sizes, int n_in,
                              void* d_out, int out_size, void* d_ws, size_t ws_size,
                              hipStream_t stream) {
    const float* img  = (const float*)d_in[0];   // input1: (1,1,160,192,224) f32
    const float* disp = (const float*)d_in[1];   // input2: (1,3,160,192,224) f32
    float* out = (float*)d_out;                  // (1,1,160,192,224) f32

    // Separate plane pointers -> clean per-plane NT loads.
    const float* dDp = disp;
    const float* dHp = disp + NVOX;
    const float* dWp = disp + 2 * NVOX;

    dim3 block(56, 4, 1);                 // 224 threads = 7 wave32s
    dim3 grid(1, HH / 4, DD);             // 48 * 160 = 7680 blocks
    dense3d_warp_kernel<<<grid, block, 0, stream>>>(img, dDp, dHp, dWp, out);
}